// GCNNet_43731357008179
// MI455X (gfx1250) — compile-verified
//
#include <hip/hip_runtime.h>
#include <stdint.h>

typedef __attribute__((ext_vector_type(2))) float v2f;
typedef __attribute__((ext_vector_type(8))) float v8f;

#define FEAT 512
#define HID  16
#define LAB  40
#define LABP 48

// Async global->LDS copy of 16B: vdst = LDS byte address (low 32 bits of the
// generic shared pointer == wave-relative LDS address per ISA 10.2 aperture
// rules), vaddr = 64-bit global address. Tracked on ASYNCcnt.
__device__ __forceinline__ void async_copy_b128(uint32_t lds_byte_addr, const void* gaddr) {
    asm volatile("global_load_async_to_lds_b128 %0, %1, off"
                 :: "v"(lds_byte_addr), "v"(gaddr)
                 : "memory");
}
__device__ __forceinline__ void wait_asynccnt0() {
    asm volatile("s_wait_asynccnt 0x0" ::: "memory");
}

// ---------------- utility kernels ----------------

__global__ void fill_kernel(float* __restrict__ p, long n, float v) {
    long i = (long)blockIdx.x * blockDim.x + threadIdx.x;
    if (i < n) p[i] = v;
}

// deg[dst[e]] += 1  (deg pre-initialized to 1.0 for the self-loop)
__global__ void deg_kernel(const int* __restrict__ dst, float* __restrict__ deg, long E) {
    long e = (long)blockIdx.x * blockDim.x + threadIdx.x;
    if (e < E) atomicAdd(&deg[dst[e]], 1.0f);
}

__global__ void rsqrt_kernel(float* __restrict__ d, long n) {
    long i = (long)blockIdx.x * blockDim.x + threadIdx.x;
    if (i < n) d[i] = rsqrtf(d[i]);
}

// ---------------- GEMM1: h1 = x @ W1, [N,512]x[512,16] ----------------
// One wave per 16-row tile; N==HID==16 exactly matches the WMMA tile.
// W1 (32KB) staged in LDS via async global->LDS b128 copies (ASYNCcnt path).
// fp32 WMMA 16x16x4, 128 K-steps.
__global__ void __launch_bounds__(256) gemm1_kernel(const float* __restrict__ x,
                                                    const float* __restrict__ W,
                                                    float* __restrict__ h, int nrows) {
    __shared__ float Ws[FEAT * HID];
    {
        // 32768 bytes = 2048 x b128; 256 threads -> 8 async copies each.
        uint32_t lds_base = (uint32_t)(uintptr_t)Ws;
        const char* gbase = (const char*)W;
        uint32_t toff = threadIdx.x * 16u;
#pragma unroll
        for (int it = 0; it < 8; ++it) {
            uint32_t o = toff + (uint32_t)it * 4096u;
            async_copy_b128(lds_base + o, gbase + o);
        }
        wait_asynccnt0();
    }
    __syncthreads();

    const int wave   = threadIdx.x >> 5;
    const int lane   = threadIdx.x & 31;
    const int ntiles = (nrows + 15) >> 4;
    const int tile   = blockIdx.x * 8 + wave;
    if (tile >= ntiles) return;                 // uniform per-wave guard (EXEC all-ones inside)

    const int row0 = tile << 4;
    const int m    = lane & 15;                 // A row / B,D column
    const int koff = (lane >> 4) << 1;          // 0 or 2

    const float2* xr = (const float2*)(x + (size_t)(row0 + m) * FEAT + koff);
    v8f acc = {};
#pragma unroll 4
    for (int k = 0; k < FEAT; k += 4) {
        float2 a2 = xr[k >> 1];                 // A[m][k+koff], A[m][k+koff+1]
        v2f a; a.x = a2.x; a.y = a2.y;
        v2f b;                                  // B[k+koff][m], B[k+koff+1][m]
        b.x = Ws[(k + koff) * HID + m];
        b.y = Ws[(k + koff + 1) * HID + m];
        acc = __builtin_amdgcn_wmma_f32_16x16x4_f32(false, a, false, b,
                                                    (short)0, acc, false, false);
    }
    const int mrow = (lane >> 4) << 3;          // D: VGPR v -> row v (+8 for hi lanes)
#pragma unroll
    for (int v = 0; v < 8; v++)
        h[(size_t)(row0 + mrow + v) * HID + m] = acc[v];
}

// ---------------- agg1: agg[dst] += h1[src] * dis[src]*dis[dst] ----------------
// 16 lanes per edge -> coalesced 64B gather per edge, scalar f32 atomics scatter.
__global__ void agg1_kernel(const int* __restrict__ src, const int* __restrict__ dst,
                            const float* __restrict__ dis, const float* __restrict__ h1,
                            float* __restrict__ agg, long E) {
    long t = (long)blockIdx.x * blockDim.x + threadIdx.x;
    long e = t >> 4;
    int  f = (int)(t & 15);
    if (e < E) {
        int s = src[e], d = dst[e];
        float nrm = dis[s] * dis[d];
        atomicAdd(&agg[(size_t)d * HID + f], h1[(size_t)s * HID + f] * nrm);
    }
}

// h1p = relu(agg1 + self_loop(h1*dis^2) + b1)
__global__ void relu_kernel(const float* __restrict__ agg, const float* __restrict__ h1,
                            const float* __restrict__ dis, const float* __restrict__ b,
                            float* __restrict__ out, long n) {
    long t = (long)blockIdx.x * blockDim.x + threadIdx.x;
    if (t < n) {
        long i = t >> 4;
        int  f = (int)(t & 15);
        float di = dis[i];
        float v = agg[t] + h1[t] * di * di + b[f];
        out[t] = v > 0.0f ? v : 0.0f;
    }
}

// ---------------- GEMM2: h2 = h1p @ W2, [N,16]x[16,40] (N padded to 48) ----------
// W2 rows (160B each) staged via async b128 copies into the padded LDS layout;
// pad columns zeroed with direct LDS stores.
__global__ void __launch_bounds__(256) gemm2_kernel(const float* __restrict__ hin,
                                                    const float* __restrict__ W,
                                                    float* __restrict__ hout, int nrows) {
    __shared__ float Ws[HID * LABP];
    {
        uint32_t lds_base = (uint32_t)(uintptr_t)Ws;
        int t = threadIdx.x;
        if (t < HID * (LAB / 4)) {              // 160 b128 copies: row k, quad q
            int k = t / (LAB / 4), q = t % (LAB / 4);
            async_copy_b128(lds_base + (uint32_t)(k * LABP + q * 4) * 4u,
                            W + (size_t)k * LAB + q * 4);
        }
        int p = t - HID * (LAB / 4);            // 128 pad elements: row k, col 40..47
        if (p >= 0 && p < HID * (LABP - LAB)) {
            int k = p / (LABP - LAB), n = LAB + p % (LABP - LAB);
            Ws[k * LABP + n] = 0.0f;
        }
        wait_asynccnt0();
    }
    __syncthreads();

    const int wave   = threadIdx.x >> 5;
    const int lane   = threadIdx.x & 31;
    const int ntiles = (nrows + 15) >> 4;
    const int tile   = blockIdx.x * 8 + wave;
    if (tile >= ntiles) return;

    const int row0 = tile << 4;
    const int m    = lane & 15;
    const int koff = (lane >> 4) << 1;

    const float2* ar = (const float2*)(hin + (size_t)(row0 + m) * HID + koff);
    v8f acc0 = {}, acc1 = {}, acc2 = {};
#pragma unroll
    for (int k = 0; k < HID; k += 4) {
        float2 a2 = ar[k >> 1];
        v2f a; a.x = a2.x; a.y = a2.y;
        v2f b0, b1, b2;
        b0.x = Ws[(k + koff) * LABP + m];       b0.y = Ws[(k + koff + 1) * LABP + m];
        b1.x = Ws[(k + koff) * LABP + 16 + m];  b1.y = Ws[(k + koff + 1) * LABP + 16 + m];
        b2.x = Ws[(k + koff) * LABP + 32 + m];  b2.y = Ws[(k + koff + 1) * LABP + 32 + m];
        acc0 = __builtin_amdgcn_wmma_f32_16x16x4_f32(false, a, false, b0, (short)0, acc0, false, false);
        acc1 = __builtin_amdgcn_wmma_f32_16x16x4_f32(false, a, false, b1, (short)0, acc1, false, false);
        acc2 = __builtin_amdgcn_wmma_f32_16x16x4_f32(false, a, false, b2, (short)0, acc2, false, false);
    }
    const int mrow = (lane >> 4) << 3;
#pragma unroll
    for (int v = 0; v < 8; v++) {
        size_t r = (size_t)(row0 + mrow + v) * LAB;
        hout[r + m]      = acc0[v];
        hout[r + 16 + m] = acc1[v];
        if (32 + m < LAB) hout[r + 32 + m] = acc2[v];
    }
}

// ---------------- agg2: out[dst] += h2[src]*norm, 8 features per thread ---------
__global__ void agg2_kernel(const int* __restrict__ src, const int* __restrict__ dst,
                            const float* __restrict__ dis, const float* __restrict__ h2,
                            float* __restrict__ out, long E) {
    long t = (long)blockIdx.x * blockDim.x + threadIdx.x;
    long e = t / 5;
    int  c = (int)(t % 5);                       // 5 chunks of 8 features
    if (e < E) {
        int s = src[e], d = dst[e];
        float nrm = dis[s] * dis[d];
        const float4* p = (const float4*)(h2 + (size_t)s * LAB + c * 8);  // 32B aligned
        float4 v0 = p[0], v1 = p[1];
        float* o = out + (size_t)d * LAB + c * 8;
        atomicAdd(o + 0, v0.x * nrm); atomicAdd(o + 1, v0.y * nrm);
        atomicAdd(o + 2, v0.z * nrm); atomicAdd(o + 3, v0.w * nrm);
        atomicAdd(o + 4, v1.x * nrm); atomicAdd(o + 5, v1.y * nrm);
        atomicAdd(o + 6, v1.z * nrm); atomicAdd(o + 7, v1.w * nrm);
    }
}

// ---------------- fused self-loop + bias + log_softmax (in-place on out) --------
__global__ void final_kernel(float* __restrict__ out, const float* __restrict__ h2,
                             const float* __restrict__ dis, const float* __restrict__ b,
                             long n) {
    long i = (long)blockIdx.x * blockDim.x + threadIdx.x;
    if (i >= n) return;
    float di = dis[i];
    float d2 = di * di;
    const float* o  = out + (size_t)i * LAB;
    const float* hh = h2  + (size_t)i * LAB;
    float v[LAB];
    float mx = -1e30f;
#pragma unroll
    for (int f = 0; f < LAB; f++) {
        v[f] = o[f] + hh[f] * d2 + b[f];
        mx = fmaxf(mx, v[f]);
    }
    float s = 0.0f;
#pragma unroll
    for (int f = 0; f < LAB; f++) s += __expf(v[f] - mx);
    float lse = mx + __logf(s);
    float* ow = out + (size_t)i * LAB;
#pragma unroll
    for (int f = 0; f < LAB; f++) ow[f] = v[f] - lse;
}

// ---------------- launcher ----------------

extern "C" void kernel_launch(void* const* d_in, const int* in_sizes, int n_in,
                              void* d_out, int out_size, void* d_ws, size_t ws_size,
                              hipStream_t stream) {
    const float* x  = (const float*)d_in[0];
    const int*   ei = (const int*)d_in[1];
    const float* W1 = (const float*)d_in[2];
    const float* b1 = (const float*)d_in[3];
    const float* W2 = (const float*)d_in[4];
    const float* b2 = (const float*)d_in[5];
    float* out = (float*)d_out;

    const long N = in_sizes[0] / FEAT;   // 100000
    const long E = in_sizes[1] / 2;      // 3200000
    const int* src = ei;
    const int* dst = ei + E;

    // workspace partition (floats): dis | h1 | agg1 | h1p | h2
    float* ws   = (float*)d_ws;
    long   Npad = ((N + 63) / 64) * 64;
    float* dis  = ws;                    // deg -> rsqrt in place
    float* h1   = dis + Npad;            // N*16
    float* agg1 = h1 + N * HID;          // N*16
    float* h1p  = agg1 + N * HID;        // N*16
    float* h2   = h1p + N * HID;         // N*40

    dim3 B(256);
    auto G = [](long n) { return dim3((unsigned)((n + 255) / 256)); };

    // zero/init accumulators (re-done every call: deterministic under graph replay)
    fill_kernel<<<G(N), B, 0, stream>>>(dis, N, 1.0f);                 // self-loop degree
    fill_kernel<<<G(N * HID), B, 0, stream>>>(agg1, N * HID, 0.0f);
    fill_kernel<<<G(N * LAB), B, 0, stream>>>(out, N * LAB, 0.0f);

    deg_kernel<<<G(E), B, 0, stream>>>(dst, dis, E);
    rsqrt_kernel<<<G(N), B, 0, stream>>>(dis, N);

    long ntiles = (N + 15) / 16;
    dim3 Gg((unsigned)((ntiles + 7) / 8));
    gemm1_kernel<<<Gg, B, 0, stream>>>(x, W1, h1, (int)N);
    agg1_kernel<<<G(E * 16), B, 0, stream>>>(src, dst, dis, h1, agg1, E);
    relu_kernel<<<G(N * HID), B, 0, stream>>>(agg1, h1, dis, b1, h1p, N * HID);
    gemm2_kernel<<<Gg, B, 0, stream>>>(h1p, W2, h2, (int)N);
    agg2_kernel<<<G(E * 5), B, 0, stream>>>(src, dst, dis, h2, out, E);
    final_kernel<<<G(N), B, 0, stream>>>(out, h2, dis, b2, N);
}